// IndRNN_88570815578223
// MI455X (gfx1250) — compile-verified
//
#include <hip/hip_runtime.h>

#define BB 32
#define TQ 1024
#define FF 256
#define HH 512
#define HP (HH + 4)   // padded LDS row stride (516: 4-bank lane skew, 16B-aligned rows)

typedef __attribute__((ext_vector_type(2))) float v2f;
typedef __attribute__((ext_vector_type(8))) float v8f;

// clamp value = 2^(1/1024)
#define CLAMP_V 1.0006771307f

// ---------------------------------------------------------------------------
// Kernel 1: xh[b*T+t][h] = sum_f x[b*T+t][f] * W_in0[h][f] + b_in0[h]
// Grid: (H/64, (B*T)/(16*8)); block 256 (8 waves). Each wave: 16x64 strip.
// ---------------------------------------------------------------------------
__global__ __launch_bounds__(256) void indrnn_xh_gemm(
    const float* __restrict__ x,
    const float* __restrict__ W0,
    const float* __restrict__ bin0,
    float* __restrict__ xh)
{
    const int lane = threadIdx.x & 31;
    const int wave = threadIdx.x >> 5;
    const int m0   = (blockIdx.y * 8 + wave) * 16;   // row tile (over B*T)
    const int n0   = blockIdx.x * 64;                // col strip (over H)

    const int mRow = lane & 15;
    const int nIdx = lane & 15;
    const int kb   = (lane >> 4) << 1;               // K pair base: 0 or 2

    const float* aBase = x + (size_t)(m0 + mRow) * FF + kb;

    v8f acc[4] = {};
    for (int k0 = 0; k0 < FF; k0 += 4) {
        v2f a = *(const v2f*)(aBase + k0);
#pragma unroll
        for (int j = 0; j < 4; ++j) {
            const float* bPtr = W0 + (size_t)(n0 + j * 16 + nIdx) * FF + k0 + kb;
            v2f b = *(const v2f*)bPtr;
            acc[j] = __builtin_amdgcn_wmma_f32_16x16x4_f32(
                false, a, false, b, (short)0, acc[j], false, false);
        }
    }

    const int rowHalf = lane >> 4;
#pragma unroll
    for (int j = 0; j < 4; ++j) {
        const int n = n0 + j * 16 + nIdx;
        const float bias = bin0[n];
#pragma unroll
        for (int i = 0; i < 8; ++i) {
            const int m = m0 + rowHalf * 8 + i;
            xh[(size_t)m * HH + n] = acc[j][i] + bias;
        }
    }
}

// ---------------------------------------------------------------------------
// Kernel 2: sequential scan. Grid: (H/64, B/16); block 128 (4 waves).
// Per-block LDS working set (~169 KB of the WGP's 320 KB):
//   s0   : full recurrent state for 16 batch rows (redundant across N-slices)
//   wl   : this block's 64-column slice of W_in1, loaded once
//   w0l/b0l : pre-clamped elementwise recurrence params
// Each wave owns one 16x16 s1 subtile in registers across all 1024 steps;
// the per-step GEMM runs entirely out of LDS (ds_load_b64 -> v_wmma f32).
// ---------------------------------------------------------------------------
__global__ __launch_bounds__(128) void indrnn_scan(
    const float* __restrict__ xh,
    const float* __restrict__ state,
    const float* __restrict__ W1,
    const float* __restrict__ bin1,
    const float* __restrict__ w0,
    const float* __restrict__ b0v,
    const float* __restrict__ w1,
    const float* __restrict__ b1v,
    float* __restrict__ outs,     // (B,T,H)
    float* __restrict__ fstate)   // (B,2,H)
{
    __shared__ float s0[16][HP];
    __shared__ float wl[64][HP];
    __shared__ float w0l[HH];
    __shared__ float b0l[HH];

    const int tid  = threadIdx.x;
    const int lane = tid & 31;
    const int wave = tid >> 5;                 // 0..3
    const int b0i  = blockIdx.y * 16;          // batch row offset
    const int nBase = blockIdx.x * 64;         // this block's H slice
    const int n0    = nBase + wave * 16;       // this wave's 16 columns

    // init s0 from state[:,0,:]
    for (int idx = tid; idx < 16 * HH; idx += 128) {
        const int m = idx >> 9, h = idx & (HH - 1);
        s0[m][h] = state[((size_t)(b0i + m) * 2 + 0) * HH + h];
    }
    // stage W_in1 slice (64 rows x 512) into LDS once, float4 loads
    for (int idx = tid; idx < 64 * (HH / 4); idx += 128) {
        const int r = idx / (HH / 4), c4 = (idx % (HH / 4)) * 4;
        *(float4*)&wl[r][c4] = *(const float4*)&W1[(size_t)(nBase + r) * HH + c4];
    }
    // pre-clamped elementwise params
    for (int h = tid; h < HH; h += 128) {
        w0l[h] = fminf(fmaxf(w0[h], -CLAMP_V), CLAMP_V);
        b0l[h] = b0v[h];
    }

    // per-lane column constants for this wave's 16 columns
    const int nIdx   = lane & 15;
    const int nCol   = n0 + nIdx;
    const float w1c  = fminf(fmaxf(w1[nCol], -CLAMP_V), CLAMP_V);
    const float bi1c = bin1[nCol];
    const float b1c  = b1v[nCol];

    const int rowHalf = lane >> 4;
    float s1r[8];
#pragma unroll
    for (int i = 0; i < 8; ++i)
        s1r[i] = state[((size_t)(b0i + rowHalf * 8 + i) * 2 + 1) * HH + nCol];

    __syncthreads();

    const int kb   = (lane >> 4) << 1;        // K pair base: 0 or 2
    const int mRow = lane & 15;
    const int rL   = wave * 16 + nIdx;        // local W row for B-fragment

    for (int t = 0; t < TQ; ++t) {
        // elementwise s0 update over full H
        for (int idx = tid; idx < 16 * HH; idx += 128) {
            const int m = idx >> 9, h = idx & (HH - 1);
            const float xv = xh[((size_t)(b0i + m) * TQ + t) * HH + h];
            float v = xv + s0[m][h] * w0l[h] + b0l[h];
            s0[m][h] = v > 0.f ? v : 0.f;
        }
        // prefetch next step's xh slice (32 KB -> 256 cachelines / block)
        if (t + 1 < TQ) {
            for (int idx = tid; idx < 256; idx += 128) {
                const int m = idx >> 4, c = idx & 15;
                __builtin_prefetch(
                    &xh[((size_t)(b0i + m) * TQ + (t + 1)) * HH + c * 32], 0, 3);
            }
        }
        __syncthreads();

        // GEMM: acc[m][n] = sum_k s0[m][k] * W_in1[n][k], K=512,
        // two interleaved accumulator chains, all operands from LDS
        v8f acc0 = {}, acc1 = {};
        for (int k0 = 0; k0 < HH; k0 += 8) {
            v2f a0 = *(const v2f*)&s0[mRow][k0 + kb];
            v2f bb0 = *(const v2f*)&wl[rL][k0 + kb];
            acc0 = __builtin_amdgcn_wmma_f32_16x16x4_f32(
                false, a0, false, bb0, (short)0, acc0, false, false);
            v2f a1 = *(const v2f*)&s0[mRow][k0 + 4 + kb];
            v2f bb1 = *(const v2f*)&wl[rL][k0 + 4 + kb];
            acc1 = __builtin_amdgcn_wmma_f32_16x16x4_f32(
                false, a1, false, bb1, (short)0, acc1, false, false);
        }

        // s1 update + output store
#pragma unroll
        for (int i = 0; i < 8; ++i) {
            float v = acc0[i] + acc1[i] + bi1c + s1r[i] * w1c + b1c;
            v = v > 0.f ? v : 0.f;
            s1r[i] = v;
            const int m = b0i + rowHalf * 8 + i;
            outs[((size_t)m * TQ + t) * HH + nCol] = v;
        }
        __syncthreads();   // protect s0 before next step's overwrite
    }

    // final state: s0 written once (slice-0 blocks only), s1 from registers
    if (blockIdx.x == 0) {
        for (int idx = tid; idx < 16 * HH; idx += 128) {
            const int m = idx >> 9, h = idx & (HH - 1);
            fstate[((size_t)(b0i + m) * 2 + 0) * HH + h] = s0[m][h];
        }
    }
#pragma unroll
    for (int i = 0; i < 8; ++i) {
        const int m = b0i + rowHalf * 8 + i;
        fstate[((size_t)m * 2 + 1) * HH + nCol] = s1r[i];
    }
}

// ---------------------------------------------------------------------------
extern "C" void kernel_launch(void* const* d_in, const int* in_sizes, int n_in,
                              void* d_out, int out_size, void* d_ws, size_t ws_size,
                              hipStream_t stream) {
    const float* x     = (const float*)d_in[0];
    const float* state = (const float*)d_in[1];
    const float* W_in0 = (const float*)d_in[2];
    const float* b_in0 = (const float*)d_in[3];
    const float* W_in1 = (const float*)d_in[4];
    const float* b_in1 = (const float*)d_in[5];
    const float* w0    = (const float*)d_in[6];
    const float* b0    = (const float*)d_in[7];
    const float* w1    = (const float*)d_in[8];
    const float* b1    = (const float*)d_in[9];

    float* outs   = (float*)d_out;                       // B*T*H
    float* fstate = outs + (size_t)BB * TQ * HH;         // B*2*H
    float* xh     = (float*)d_ws;                        // B*T*H scratch

    // Kernel 1: input projection GEMM (f32 WMMA)
    dim3 g1(HH / 64, (BB * TQ) / (16 * 8));
    indrnn_xh_gemm<<<g1, 256, 0, stream>>>(x, W_in0, b_in0, xh);

    // Kernel 2: recurrent scan (LDS-resident weights, f32 WMMA per step)
    dim3 g2(HH / 64, BB / 16);
    indrnn_scan<<<g2, 128, 0, stream>>>(xh, state, W_in1, b_in1,
                                        w0, b0, w1, b1, outs, fstate);
}